// Model_31731218383103
// MI455X (gfx1250) — compile-verified
//
#include <hip/hip_runtime.h>
#include <math.h>

typedef __attribute__((ext_vector_type(2))) float v2f;
typedef __attribute__((ext_vector_type(8))) float v8f;

#define EPS_NOISE 0.1f
#define PERM_A 48271u

__device__ __forceinline__ float pick8(v8f v, int i) {
  switch (i & 7) {
    case 0: return v[0];
    case 1: return v[1];
    case 2: return v[2];
    case 3: return v[3];
    case 4: return v[4];
    case 5: return v[5];
    case 6: return v[6];
    default: return v[7];
  }
}

// ---------------------------------------------------------------------------
// Edge scoring: s[e] = threshold( (cos(u_t, i_t)+1)/2 , beta )
// 16 edges per wave; three f32 WMMA Gram chains give |u|^2, |i|^2, u.i exactly.
// A(16x4) and B(4x16) f32 VGPR layouts coincide per-lane:
//   lane L supplies float2 of row (L&15) at col 4t + 2*(L>>4).
// ---------------------------------------------------------------------------
__global__ void edge_score_kernel(const int* __restrict__ u_id, const int* __restrict__ i_id,
                                  const int* __restrict__ T, const float* __restrict__ beta_p,
                                  const float* __restrict__ user_emb,
                                  const float* __restrict__ item_emb,
                                  const float* __restrict__ W0, const float* __restrict__ W1,
                                  float* __restrict__ s_out, int E) {
  __shared__ float sW0[24 * 32];
  __shared__ float sW1[7 * 32];
  for (int k = threadIdx.x; k < 24 * 32; k += blockDim.x) sW0[k] = W0[k];
  for (int k = threadIdx.x; k < 7 * 32; k += blockDim.x) sW1[k] = W1[k];
  __syncthreads();

  const float beta = beta_p[0];
  const int lane = threadIdx.x & 31;
  const int wavesPerBlock = blockDim.x >> 5;
  const int wave = blockIdx.x * wavesPerBlock + (threadIdx.x >> 5);
  const int nwaves = gridDim.x * wavesPerBlock;
  const int ntile = E >> 4;

  for (int tile = wave; tile < ntile; tile += nwaves) {
    const int ebase = tile << 4;
    const int m = lane & 15;            // edge within tile served by this lane
    const int e = ebase + m;
    const int uid = u_id[e];
    const int iid = i_id[e];
    const int t0 = T[2 * e + 0];
    const int t1 = T[2 * e + 1];
    const int k0 = (lane >> 4) << 1;    // 0 (lanes 0-15) or 2 (lanes 16-31)
    const float* __restrict__ up = user_emb + (size_t)uid * 64;
    const float* __restrict__ ip = item_emb + (size_t)iid * 64;

    v8f guu = {}; v8f gii = {}; v8f gui = {};
#pragma unroll
    for (int t = 0; t < 16; ++t) {
      const int d = 4 * t + k0;         // even; d and d+1 in same half (32 is even)
      float te0, te1;
      if (d < 32) { te0 = sW0[t0 * 32 + d];      te1 = sW0[t0 * 32 + d + 1]; }
      else        { te0 = sW1[t1 * 32 + d - 32]; te1 = sW1[t1 * 32 + d - 31]; }
      v2f a, b;
      a.x = up[d] + te0;  a.y = up[d + 1] + te1;
      b.x = ip[d] + te0;  b.y = ip[d + 1] + te1;
      guu = __builtin_amdgcn_wmma_f32_16x16x4_f32(false, a, false, a, (short)0, guu, false, false);
      gii = __builtin_amdgcn_wmma_f32_16x16x4_f32(false, b, false, b, (short)0, gii, false, false);
      gui = __builtin_amdgcn_wmma_f32_16x16x4_f32(false, a, false, b, (short)0, gui, false, false);
    }
    // Diagonal of 16x16 f32 C/D: M = vgpr + 8*(lane>=16), N = lane&15.
    // diag[m] (m<8)  -> lane m,    c[m]
    // diag[m] (m>=8) -> lane m+16, c[m-8]
    if ((lane < 8) || (lane >= 24)) {
      const int ci = (lane < 8) ? lane : (lane - 24);
      const int em = (lane < 8) ? lane : (lane - 16);
      const float duu = pick8(guu, ci);
      const float dii = pick8(gii, ci);
      const float dui = pick8(gui, ci);
      float sv = dui / (fmaxf(sqrtf(duu), 1e-12f) * fmaxf(sqrtf(dii), 1e-12f));
      sv = (sv + 1.0f) * 0.5f;
      s_out[ebase + em] = (sv < beta) ? 0.0f : sv;
    }
  }

  // Scalar tail for E % 16 edges (rare; E=1M is tile-exact).
  const int tail0 = ntile << 4;
  const int gtid = blockIdx.x * blockDim.x + threadIdx.x;
  const int gsize = gridDim.x * blockDim.x;
  for (int e = tail0 + gtid; e < E; e += gsize) {
    const int uid = u_id[e], iid = i_id[e];
    const int t0 = T[2 * e], t1 = T[2 * e + 1];
    const float* up = user_emb + (size_t)uid * 64;
    const float* ip = item_emb + (size_t)iid * 64;
    float su = 0.f, si = 0.f, sui = 0.f;
    for (int d = 0; d < 64; ++d) {
      const float te = (d < 32) ? sW0[t0 * 32 + d] : sW1[t1 * 32 + d - 32];
      const float uu = up[d] + te, ii = ip[d] + te;
      su += uu * uu; si += ii * ii; sui += uu * ii;
    }
    float sv = sui / (fmaxf(sqrtf(su), 1e-12f) * fmaxf(sqrtf(si), 1e-12f));
    sv = (sv + 1.0f) * 0.5f;
    s_out[e] = (sv < beta) ? 0.0f : sv;
  }
}

// ---------------------------------------------------------------------------
// Fused dual SpMM: u_next += A i_cur and i_next += A^T u_cur in one edge sweep.
// Thread per (edge, float4-chunk); tables are L2-resident (77MB < 192MB L2).
// ---------------------------------------------------------------------------
__global__ void spmm_both_kernel(const int* __restrict__ u_id, const int* __restrict__ i_id,
                                 const float* __restrict__ s,
                                 const float* __restrict__ u_cur, const float* __restrict__ i_cur,
                                 float* __restrict__ u_next, float* __restrict__ i_next, int E) {
  const long tid = (long)blockIdx.x * blockDim.x + threadIdx.x;
  if (tid >= (long)E * 16) return;
  const int e = (int)(tid >> 4);
  const int c = (int)(tid & 15);
  const float sv = s[e];
  if (sv == 0.0f) return;
  const int uid = u_id[e], iid = i_id[e];
  const float4 x = *(const float4*)(i_cur + (size_t)iid * 64 + c * 4);
  const float4 y = *(const float4*)(u_cur + (size_t)uid * 64 + c * 4);
  float* du = u_next + (size_t)uid * 64 + c * 4;
  float* di = i_next + (size_t)iid * 64 + c * 4;
  atomicAdd(du + 0, sv * x.x); atomicAdd(du + 1, sv * x.y);
  atomicAdd(du + 2, sv * x.z); atomicAdd(du + 3, sv * x.w);
  atomicAdd(di + 0, sv * y.x); atomicAdd(di + 1, sv * y.y);
  atomicAdd(di + 2, sv * y.z); atomicAdd(di + 3, sv * y.w);
}

// Wave-per-row L2 norm (lane loads float2 -> 256B coalesced per row).
__global__ void rownorm_kernel(const float* __restrict__ X, float* __restrict__ norms, int rows) {
  const int lane = threadIdx.x & 31;
  const int row = blockIdx.x * (blockDim.x >> 5) + (threadIdx.x >> 5);
  if (row >= rows) return;
  const float2 v = *(const float2*)(X + (size_t)row * 64 + lane * 2);
  float ss = v.x * v.x + v.y * v.y;
#pragma unroll
  for (int off = 16; off > 0; off >>= 1) ss += __shfl_xor(ss, off, 32);
  if (lane == 0) norms[row] = sqrtf(ss);
}

// dst[r] = src[r] + EPS * src[perm(r)] / max(|src[perm(r)]|, 1e-12); acc += dst
__global__ void perturb_acc_kernel(const float* __restrict__ src, const float* __restrict__ norms,
                                   float* __restrict__ dst, float* __restrict__ acc,
                                   int rows, unsigned bseed) {
  const long tid = (long)blockIdx.x * blockDim.x + threadIdx.x;
  if (tid >= (long)rows * 16) return;
  const int r = (int)(tid >> 4);
  const int c = (int)(tid & 15);
  const unsigned p =
      (unsigned)(((unsigned long long)PERM_A * (unsigned)r + bseed) % (unsigned)rows);
  const float4 x = *(const float4*)(src + (size_t)r * 64 + c * 4);
  const float4 z = *(const float4*)(src + (size_t)p * 64 + c * 4);
  const float inv = EPS_NOISE / fmaxf(norms[p], 1e-12f);
  float4 d;
  d.x = x.x + z.x * inv; d.y = x.y + z.y * inv;
  d.z = x.z + z.z * inv; d.w = x.w + z.w * inv;
  *(float4*)(dst + (size_t)r * 64 + c * 4) = d;
  float4* ap = (float4*)(acc + (size_t)r * 64 + c * 4);
  float4 a = *ap;
  a.x += d.x; a.y += d.y; a.z += d.z; a.w += d.w;
  *ap = a;
}

__global__ void accum_kernel(float* __restrict__ acc, const float* __restrict__ src, long n4) {
  const long tid = (long)blockIdx.x * blockDim.x + threadIdx.x;
  if (tid >= n4) return;
  float4 a = ((float4*)acc)[tid];
  const float4 v = ((const float4*)src)[tid];
  a.x += v.x; a.y += v.y; a.z += v.z; a.w += v.w;
  ((float4*)acc)[tid] = a;
}

__global__ void scale_kernel(float* __restrict__ out, long n, float sc) {
  const long tid = (long)blockIdx.x * blockDim.x + threadIdx.x;
  if (tid < n) out[tid] *= sc;
}

extern "C" void kernel_launch(void* const* d_in, const int* in_sizes, int n_in,
                              void* d_out, int out_size, void* d_ws, size_t ws_size,
                              hipStream_t stream) {
  const int* u_id = (const int*)d_in[0];
  const int* i_id = (const int*)d_in[1];
  const int* T = (const int*)d_in[2];
  const float* beta = (const float*)d_in[3];
  const float* user_emb = (const float*)d_in[4];
  const float* item_emb = (const float*)d_in[5];
  const float* W0 = (const float*)d_in[6];
  const float* W1 = (const float*)d_in[7];
  const int E = in_sizes[0];
  const int N = in_sizes[4] / 64;
  const int M = in_sizes[5] / 64;
  const size_t N64 = (size_t)N * 64, M64 = (size_t)M * 64;

  float* ws = (float*)d_ws;
  float* s = ws;
  float* uA = s + E;
  float* uB = uA + N64;
  float* iA = uB + N64;
  float* iB = iA + M64;
  float* nU = iB + M64;
  float* nI = nU + N;

  float* out = (float*)d_out;
  float* o_uf = out;
  float* o_if = o_uf + N64;
  float* o_u1 = o_if + M64;
  float* o_i1 = o_u1 + N64;
  float* o_u2 = o_i1 + M64;
  float* o_i2 = o_u2 + N64;

  const int TB = 256;

  // 1) edge scores (shared by all passes)
  edge_score_kernel<<<1024, TB, 0, stream>>>(u_id, i_id, T, beta, user_emb, item_emb,
                                             W0, W1, s, E);

  // 2) zero all outputs (they are accumulators)
  hipMemsetAsync(d_out, 0, (size_t)out_size * sizeof(float), stream);

  const long spmmT = (long)E * 16;
  const int spmmB = (int)((spmmT + TB - 1) / TB);

  auto launch_spmm = [&](const float* uc, const float* ic, float* un, float* in_) {
    hipMemsetAsync(un, 0, N64 * sizeof(float), stream);
    hipMemsetAsync(in_, 0, M64 * sizeof(float), stream);
    spmm_both_kernel<<<spmmB, TB, 0, stream>>>(u_id, i_id, s, uc, ic, un, in_, E);
  };

  // 3) clean pass: u_{l+1} = A i_l ; i_{l+1} = A^T u_l ; acc += new
  {
    const float* uc = user_emb;
    const float* ic = item_emb;
    float* ubuf[2] = {uA, uB};
    float* ibuf[2] = {iA, iB};
    for (int l = 0; l < 3; ++l) {
      float* un = ubuf[l & 1];
      float* in_ = ibuf[l & 1];
      launch_spmm(uc, ic, un, in_);
      accum_kernel<<<(int)((N64 / 4 + TB - 1) / TB), TB, 0, stream>>>(o_uf, un, (long)(N64 / 4));
      accum_kernel<<<(int)((M64 / 4 + TB - 1) / TB), TB, 0, stream>>>(o_if, in_, (long)(M64 / 4));
      uc = un;
      ic = in_;
    }
  }

  // 4) perturbed passes: new = spmm(...); new += EPS*l2norm(new[perm]); acc += new
  for (int pass = 0; pass < 2; ++pass) {
    float* au = pass ? o_u2 : o_u1;
    float* ai = pass ? o_i2 : o_i1;
    const float* uc = user_emb;
    const float* ic = item_emb;
    for (int l = 0; l < 3; ++l) {
      launch_spmm(uc, ic, uA, iA);
      rownorm_kernel<<<(N + 7) / 8, TB, 0, stream>>>(uA, nU, N);
      rownorm_kernel<<<(M + 7) / 8, TB, 0, stream>>>(iA, nI, M);
      const unsigned bu = 2654435761u * (unsigned)(pass + 1) + 97u * (unsigned)l + 13u;
      const unsigned bi = 2654435761u * (unsigned)(pass + 1) + 97u * (unsigned)l + 131u;
      perturb_acc_kernel<<<(int)(((long)N * 16 + TB - 1) / TB), TB, 0, stream>>>(uA, nU, uB, au,
                                                                                 N, bu);
      perturb_acc_kernel<<<(int)(((long)M * 16 + TB - 1) / TB), TB, 0, stream>>>(iA, nI, iB, ai,
                                                                                 M, bi);
      uc = uB;
      ic = iB;
    }
  }

  // 5) mean over the 3 layers
  scale_kernel<<<(int)(((long)out_size + TB - 1) / TB), TB, 0, stream>>>(out, (long)out_size,
                                                                         1.0f / 3.0f);
}